// CommNetMLP_17403207483511
// MI455X (gfx1250) — compile-verified
//
#include <hip/hip_runtime.h>
#include <hip/hip_bf16.h>

typedef __attribute__((ext_vector_type(2))) float v2f;
typedef __attribute__((ext_vector_type(8))) float v8f;

#define T_NA 100
#define T_NN 101
#define T_FEAT 113
#define T_HID 1024
#define GAT_ALPHA 0.2f

// big-GEMM tiling
#define BM 112      // 7 * 16 rows (covers M=100)
#define MT 7        // M tiles per wave
#define BK 16       // staged K slice
#define LDA 18      // padded LDS row stride (floats) -> conflict-free b64 reads

// ---------------------------------------------------------------------------
// Small/medium WMMA f32 GEMM: one wave per 16x16 C tile, grid.y tiles M.
// C[M,N] = act( A[M,K] @ B(^T) + bias + res ), rowScale fuses h*comm_action.
// Boundaries are clamped, never predicated: garbage rows/cols are never stored.
// ---------------------------------------------------------------------------
__global__ __launch_bounds__(256)
void gemm_wmma(const float* __restrict__ A, const float* __restrict__ B,
               const float* __restrict__ bias, const float* __restrict__ res,
               const float* __restrict__ rowScale, float* __restrict__ C,
               int M, int N, int K, int transB, int act)
{
    const int wave = threadIdx.x >> 5;
    const int lane = threadIdx.x & 31;
    const int nbase = (blockIdx.x * 8 + wave) * 16;
    const int mbase = blockIdx.y * 16;
    if (nbase >= N) return;                 // uniform per wave -> EXEC stays full

    const int hi  = lane >> 4;              // 0: K pair {0,1}, 1: K pair {2,3}
    const int l15 = lane & 15;

    int arow = mbase + l15; if (arow >= M) arow = M - 1;   // clamp (garbage ok)
    int bcol = nbase + l15; if (bcol >= N) bcol = N - 1;   // clamp (garbage ok)

    float rs = 1.0f;
    if (rowScale) rs = rowScale[arow];

    const float* Arow = A + (size_t)arow * (size_t)K + hi * 2;
    v8f acc = {};

    if (!transB) {
        const float* Bp = B + (size_t)hi * 2 * N + bcol;
#pragma unroll 4
        for (int k0 = 0; k0 < K; k0 += 4) {
            v2f a = *(const v2f*)(Arow + k0);
            a.x *= rs; a.y *= rs;
            v2f b;
            b.x = Bp[(size_t)k0 * N];
            b.y = Bp[(size_t)k0 * N + N];
            __builtin_prefetch(Bp + (size_t)(k0 + 16) * N, 0, 1);
            acc = __builtin_amdgcn_wmma_f32_16x16x4_f32(
                      false, a, false, b, (short)0, acc, false, false);
        }
    } else {
        const float* Brow = B + (size_t)bcol * (size_t)K + hi * 2;
#pragma unroll 4
        for (int k0 = 0; k0 < K; k0 += 4) {
            v2f a = *(const v2f*)(Arow + k0);
            a.x *= rs; a.y *= rs;
            v2f b = *(const v2f*)(Brow + k0);
            acc = __builtin_amdgcn_wmma_f32_16x16x4_f32(
                      false, a, false, b, (short)0, acc, false, false);
        }
    }

    const int ccol = nbase + l15;
    if (ccol < N) {
        const int crow0 = mbase + hi * 8;   // C layout: VGPR v -> M = v (+8 hi lanes)
#pragma unroll
        for (int v = 0; v < 8; ++v) {
            const int r = crow0 + v;
            if (r < M) {
                float val = acc[v];
                if (bias) val += bias[ccol];
                if (res)  val += res[(size_t)r * N + ccol];
                if (act == 1)      val = fmaxf(val, 0.0f);
                else if (act == 2) val = (val > 0.0f) ? val : 0.01f * val;
                C[(size_t)r * N + ccol] = val;
            }
        }
    }
}

// ---------------------------------------------------------------------------
// Big-N WMMA f32 GEMM (N >= 4096), M <= 112: one block owns a 128-col N-strip
// and ALL M rows. A panel is staged in double-buffered LDS (next stage's
// global loads issued before the WMMA loop so they overlap compute), and each
// wave runs 7 WMMAs (one per 16-row M tile) per B load. B is therefore read
// from L2/HBM exactly once — removes the 7x L2 amplification of the naive
// mapping, which would otherwise exceed L2 bandwidth on the 33-47MB weights.
// Uniform control flow: every thread participates in barriers, clamped tiles
// compute garbage that is never stored, EXEC stays all-ones for WMMA.
// ---------------------------------------------------------------------------
__global__ __launch_bounds__(256)
void gemm_wmma_big(const float* __restrict__ A, const float* __restrict__ B,
                   const float* __restrict__ bias, const float* __restrict__ res,
                   const float* __restrict__ rowScale, float* __restrict__ C,
                   int M, int N, int K, int transB, int act)
{
    __shared__ float sA[2][BM * LDA];

    const int tid  = threadIdx.x;
    const int wave = tid >> 5;
    const int lane = tid & 31;
    const int hi   = lane >> 4;
    const int l15  = lane & 15;

    const int nbase = (blockIdx.x * 8 + wave) * 16;
    int bcol = nbase + l15; if (bcol >= N) bcol = N - 1;      // clamp (garbage ok)

    const int nk = (K + BK - 1) / BK;

    // staging: thread copies elements tid + i*256 of the [BM x BK] slice
    // (BM*BK = 1792 = 7*256 exactly)
    size_t aoff[7]; int ldsoff[7], kkk[7]; float rsv[7];
    #pragma unroll
    for (int i = 0; i < 7; ++i) {
        const int flat = tid + i * 256;
        const int row  = flat >> 4;          // 0..111
        const int kk   = flat & 15;
        int grow = row; if (grow >= M) grow = M - 1;
        aoff[i]   = (size_t)grow * (size_t)K + kk;
        ldsoff[i] = row * LDA + kk;
        kkk[i]    = kk;
        rsv[i]    = rowScale ? rowScale[grow] : 1.0f;
    }

    // prologue: fill stage 0
    #pragma unroll
    for (int i = 0; i < 7; ++i) {
        const int k = kkk[i];
        sA[0][ldsoff[i]] = (k < K) ? A[aoff[i]] * rsv[i] : 0.0f;
    }
    __syncthreads();

    v8f acc[MT];
    #pragma unroll
    for (int t = 0; t < MT; ++t) acc[t] = (v8f){};

    for (int ks = 0; ks < nk; ++ks) {
        const int buf = ks & 1;
        const int kbase = ks * BK;
        const bool have_next = (ks + 1 < nk);

        // issue next stage's global loads early (overlap with WMMA below)
        float ld[7];
        if (have_next) {
            const int kb2 = kbase + BK;
            #pragma unroll
            for (int i = 0; i < 7; ++i) {
                const int k = kb2 + kkk[i];
                ld[i] = (k < K) ? A[aoff[i] + (size_t)kb2] * rsv[i] : 0.0f;
            }
        }

        // compute on current stage
        #pragma unroll
        for (int kk = 0; kk < BK; kk += 4) {
            const int ke = kbase + kk + hi * 2;
            v2f b;
            if (!transB) {
                int k0c = ke;     if (k0c > K - 1) k0c = K - 1;
                int k1c = ke + 1; if (k1c > K - 1) k1c = K - 1;
                b.x = B[(size_t)k0c * N + bcol];
                b.y = B[(size_t)k1c * N + bcol];
            } else {
                int kec = ke; if (kec > K - 2) kec = K - 2;     // K%4==0 -> aligned
                b = *(const v2f*)(B + (size_t)bcol * (size_t)K + kec);
            }
            #pragma unroll
            for (int t = 0; t < MT; ++t) {
                v2f a = *(const v2f*)(&sA[buf][(t * 16 + l15) * LDA + kk + hi * 2]);
                acc[t] = __builtin_amdgcn_wmma_f32_16x16x4_f32(
                             false, a, false, b, (short)0, acc[t], false, false);
            }
        }

        if (have_next) {
            #pragma unroll
            for (int i = 0; i < 7; ++i) sA[buf ^ 1][ldsoff[i]] = ld[i];
            __syncthreads();
        }
    }

    // epilogue
    const int ccol = nbase + l15;
    if (ccol < N) {
        #pragma unroll
        for (int t = 0; t < MT; ++t) {
            const int crow0 = t * 16 + hi * 8;
            #pragma unroll
            for (int v = 0; v < 8; ++v) {
                const int r = crow0 + v;
                if (r < M) {
                    float val = acc[t][v];
                    if (bias) val += bias[ccol];
                    if (res)  val += res[(size_t)r * N + ccol];
                    if (act == 1)      val = fmaxf(val, 0.0f);
                    else if (act == 2) val = (val > 0.0f) ? val : 0.01f * val;
                    C[(size_t)r * N + ccol] = val;
                }
            }
        }
    }
}

// ---------------------------------------------------------------------------
// Per-agent GAT (irregular, <5% of FLOPs): 3 heads with masked attention +
// elu, progressive hcat@Wo accumulation, output-layer attention for the
// node_idx row only (reference discards all other rows), elu + log_softmax.
// ---------------------------------------------------------------------------
__global__ __launch_bounds__(128)
void gat_kernel(const float* __restrict__ nf,   // [A,NN,FEAT]
                const int*   __restrict__ adj,  // [A,NN,NN]
                const int*   __restrict__ nidx, // [A]
                const float* __restrict__ W,    // [3,FEAT,32]
                const float* __restrict__ av,   // [3,64]
                const float* __restrict__ Wo,   // [96,16]
                const float* __restrict__ ao,   // [32]
                float* __restrict__ enc_in)     // [A,16]
{
    __shared__ float sWh[T_NN][32];
    __shared__ float sOut[T_NN][32];
    __shared__ float sWho[T_NN][16];
    __shared__ float sS1[T_NN];
    __shared__ float sS2[T_NN];

    const int ag = blockIdx.x;
    const int t  = threadIdx.x;
    const float* x  = nf  + (size_t)ag * T_NN * T_FEAT;
    const int*   am = adj + (size_t)ag * T_NN * T_NN;

    for (int i = t; i < T_NN * 16; i += 128) (&sWho[0][0])[i] = 0.0f;

    for (int hh = 0; hh < 3; ++hh) {
        __syncthreads();
        const float* Wh = W + hh * T_FEAT * 32;
        for (int i = t; i < T_NN * 32; i += 128) {
            const int n = i >> 5, d = i & 31;
            const float* xr = x + n * T_FEAT;
            float acc = 0.0f;
            for (int f = 0; f < T_FEAT; ++f) acc += xr[f] * Wh[f * 32 + d];
            sWh[n][d] = acc;
        }
        __syncthreads();
        const float* a0 = av + hh * 64;
        for (int n = t; n < T_NN; n += 128) {
            float s1 = 0.0f, s2 = 0.0f;
            for (int d = 0; d < 32; ++d) { s1 += sWh[n][d] * a0[d]; s2 += sWh[n][d] * a0[32 + d]; }
            sS1[n] = s1; sS2[n] = s2;
        }
        __syncthreads();
        for (int n = t; n < T_NN; n += 128) {
            const float s1 = sS1[n];
            const int* mr = am + n * T_NN;
            float m = -3.4e38f;
            for (int j = 0; j < T_NN; ++j) {
                float e = s1 + sS2[j];
                e = (e > 0.0f) ? e : GAT_ALPHA * e;
                e = (mr[j] > 0) ? e : -9e15f;
                m = fmaxf(m, e);
            }
            float sum = 0.0f, o[32];
            for (int d = 0; d < 32; ++d) o[d] = 0.0f;
            for (int j = 0; j < T_NN; ++j) {
                float e = s1 + sS2[j];
                e = (e > 0.0f) ? e : GAT_ALPHA * e;
                e = (mr[j] > 0) ? e : -9e15f;
                const float w = __expf(e - m);
                sum += w;
                for (int d = 0; d < 32; ++d) o[d] += w * sWh[j][d];
            }
            const float inv = 1.0f / sum;
            for (int d = 0; d < 32; ++d) {
                const float v = o[d] * inv;
                sOut[n][d] = (v > 0.0f) ? v : (__expf(v) - 1.0f);   // elu
            }
        }
        __syncthreads();
        for (int i = t; i < T_NN * 16; i += 128) {                 // hcat @ Wo (head slice)
            const int n = i >> 4, d = i & 15;
            float acc = 0.0f;
            for (int dd = 0; dd < 32; ++dd) acc += sOut[n][dd] * Wo[(hh * 32 + dd) * 16 + d];
            sWho[n][d] += acc;
        }
    }
    __syncthreads();

    const int ni = nidx[ag];
    for (int n = t; n < T_NN; n += 128) {
        float s2 = 0.0f;
        for (int d = 0; d < 16; ++d) s2 += sWho[n][d] * ao[16 + d];
        sS2[n] = s2;
    }
    __syncthreads();
    if (t == 0) {
        float s1 = 0.0f;
        for (int d = 0; d < 16; ++d) s1 += sWho[ni][d] * ao[d];
        const int* mr = am + ni * T_NN;
        float m = -3.4e38f;
        for (int j = 0; j < T_NN; ++j) {
            float e = s1 + sS2[j];
            e = (e > 0.0f) ? e : GAT_ALPHA * e;
            e = (mr[j] > 0) ? e : -9e15f;
            m = fmaxf(m, e);
        }
        float sum = 0.0f, o[16];
        for (int d = 0; d < 16; ++d) o[d] = 0.0f;
        for (int j = 0; j < T_NN; ++j) {
            float e = s1 + sS2[j];
            e = (e > 0.0f) ? e : GAT_ALPHA * e;
            e = (mr[j] > 0) ? e : -9e15f;
            const float w = __expf(e - m);
            sum += w;
            for (int d = 0; d < 16; ++d) o[d] += w * sWho[j][d];
        }
        const float inv = 1.0f / sum;
        float mx = -3.4e38f;
        for (int d = 0; d < 16; ++d) {
            float v = o[d] * inv;
            v = (v > 0.0f) ? v : (__expf(v) - 1.0f);               // elu
            o[d] = v; mx = fmaxf(mx, v);
        }
        float se = 0.0f;
        for (int d = 0; d < 16; ++d) se += __expf(o[d] - mx);
        const float lse = mx + __logf(se);
        for (int d = 0; d < 16; ++d) enc_in[ag * 16 + d] = o[d] - lse;   // log_softmax
    }
}

// ---- comm attention: attn = softmax(q @ k^T * scale) -----------------------
__global__ __launch_bounds__(128)
void attn_kernel(const float* __restrict__ q, const float* __restrict__ k,
                 float* __restrict__ attn, int A, float scale)
{
    __shared__ float qv[16];
    __shared__ float e[128];
    __shared__ float red[2];
    const int i = blockIdx.x, t = threadIdx.x;
    if (t < 16) qv[t] = q[i * 16 + t];
    __syncthreads();
    float s = -3.4e38f;
    if (t < A) {
        float acc = 0.0f;
        for (int d = 0; d < 16; ++d) acc += qv[d] * k[t * 16 + d];
        s = acc * scale;
    }
    e[t] = s;
    __syncthreads();
    if (t == 0) {
        float m = -3.4e38f;
        for (int j = 0; j < A; ++j) m = fmaxf(m, e[j]);
        red[0] = m;
    }
    __syncthreads();
    if (t < A) e[t] = __expf(e[t] - red[0]);
    __syncthreads();
    if (t == 0) {
        float sum = 0.0f;
        for (int j = 0; j < A; ++j) sum += e[j];
        red[1] = 1.0f / sum;
    }
    __syncthreads();
    if (t < A) attn[i * A + t] = e[t] * red[1];
}

// ---- LSTM cell elementwise -------------------------------------------------
__global__ void lstm_kernel(const float* __restrict__ gates,
                            float* __restrict__ h, float* __restrict__ c,
                            int A, int H)
{
    const int idx = blockIdx.x * blockDim.x + threadIdx.x;
    if (idx >= A * H) return;
    const int m = idx / H, j = idx - m * H;
    const float* g = gates + (size_t)m * 4 * H;
    const float ig = g[j], fg = g[H + j], gg = g[2 * H + j], og = g[3 * H + j];
    const float si = 1.0f / (1.0f + __expf(-ig));
    const float sf = 1.0f / (1.0f + __expf(-fg));
    const float so = 1.0f / (1.0f + __expf(-og));
    const float cn = sf * c[idx] + si * tanhf(gg);
    c[idx] = cn;
    h[idx] = so * tanhf(cn);
}

// ---- init: h=hidden0, c=cell0, caf=float(comm_action) ----------------------
__global__ void prep_kernel(const float* __restrict__ h0, const float* __restrict__ c0,
                            const int* __restrict__ ca, float* __restrict__ h,
                            float* __restrict__ c, float* __restrict__ caf,
                            int AH, int A)
{
    const int idx = blockIdx.x * blockDim.x + threadIdx.x;
    if (idx < AH) { h[idx] = h0[idx]; c[idx] = c0[idx]; }
    if (idx < A)  caf[idx] = (float)ca[idx];
}

__global__ void copy_kernel(const float* __restrict__ src, float* __restrict__ dst, int n)
{
    const int idx = blockIdx.x * blockDim.x + threadIdx.x;
    if (idx < n) dst[idx] = src[idx];
}

// ---------------------------------------------------------------------------
static inline void launch_gemm(const float* A, const float* B, const float* bias,
                               const float* res, const float* rowScale, float* C,
                               int M, int N, int K, int transB, int act,
                               hipStream_t s)
{
    if (N >= 4096 && M <= BM) {
        dim3 g((N + 127) / 128);
        gemm_wmma_big<<<g, 256, 0, s>>>(A, B, bias, res, rowScale, C, M, N, K, transB, act);
    } else {
        dim3 g((N + 127) / 128, (M + 15) / 16);
        gemm_wmma<<<g, 256, 0, s>>>(A, B, bias, res, rowScale, C, M, N, K, transB, act);
    }
}

extern "C" void kernel_launch(void* const* d_in, const int* in_sizes, int n_in,
                              void* d_out, int out_size, void* d_ws, size_t ws_size,
                              hipStream_t stream)
{
    const float* nf       = (const float*)d_in[0];
    const float* hidden0  = (const float*)d_in[1];
    const float* cell0    = (const float*)d_in[2];
    const int*   adj      = (const int*)  d_in[3];
    const int*   node_idx = (const int*)  d_in[4];
    const int*   comm_act = (const int*)  d_in[5];
    const float* gat_W    = (const float*)d_in[6];
    const float* gat_a    = (const float*)d_in[7];
    const float* gat_Wo   = (const float*)d_in[8];
    const float* gat_ao   = (const float*)d_in[9];
    const float* enc_W1   = (const float*)d_in[10];
    const float* enc_b1   = (const float*)d_in[11];
    const float* enc_W2   = (const float*)d_in[12];
    const float* enc_b2   = (const float*)d_in[13];
    const float* Wq       = (const float*)d_in[14];
    const float* bq       = (const float*)d_in[15];
    const float* Wk       = (const float*)d_in[16];
    const float* bk       = (const float*)d_in[17];
    const float* Wv       = (const float*)d_in[18];
    const float* bv       = (const float*)d_in[19];
    const float* C_W      = (const float*)d_in[20];
    const float* C_b      = (const float*)d_in[21];
    const float* Wih      = (const float*)d_in[22];
    const float* Whh      = (const float*)d_in[23];
    const float* bih      = (const float*)d_in[24];
    const float* bhh      = (const float*)d_in[25];
    const float* map_W1   = (const float*)d_in[26];
    const float* map_b1   = (const float*)d_in[27];
    const float* map_W2   = (const float*)d_in[28];
    const float* map_b2   = (const float*)d_in[29];
    const float* gnn_W1   = (const float*)d_in[30];
    const float* gnn_b1   = (const float*)d_in[31];
    const float* gnn_W2   = (const float*)d_in[32];
    const float* gnn_b2   = (const float*)d_in[33];
    const float* vh_W     = (const float*)d_in[34];
    const float* vh_b     = (const float*)d_in[35];
    const float* vg_W     = (const float*)d_in[36];
    const float* vg_b     = (const float*)d_in[37];

    const int A = T_NA, H = T_HID;
    const int MAP_OUT = T_NN * T_FEAT;   // 11413
    const int GNN_OUT = T_NN * T_NN;     // 10201

    // workspace layout (floats; all offsets multiples of 8 for b64 loads)
    float* ws      = (float*)d_ws;
    float* enc_in  = ws;                 // 1600
    float* t1      = ws + 1600;          // 102400
    float* x       = ws + 104000;        // 102400
    float* h       = ws + 206400;        // 102400
    float* c       = ws + 308800;        // 102400
    float* caf     = ws + 411200;        // 128
    float* q       = ws + 411328;        // 1600
    float* kbuf    = ws + 412928;        // 1600
    float* v       = ws + 414528;        // 102400
    float* attn    = ws + 516928;        // 10000
    float* cmsg1   = ws + 526928;        // 102400
    float* xc      = ws + 629328;        // 102400
    float* gates   = ws + 731728;        // 409600
    float* map1    = ws + 1141328;       // 102400
    float* gnn1    = ws + 1243728;       // 102400

    // output layout (return order: h, value_head, value_global, node, edge)
    float* out_h    = (float*)d_out;                 // 102400
    float* out_vh   = out_h + A * H;                 // 100
    float* out_vg   = out_vh + A;                    // 10000
    float* out_node = out_vg + A * A;                // 1141300
    float* out_edge = out_node + A * MAP_OUT;        // 1020100

    // init h, c, comm_action float
    prep_kernel<<<(A * H + 255) / 256, 256, 0, stream>>>(hidden0, cell0, comm_act,
                                                         h, c, caf, A * H, A);

    // GAT -> enc_in[100,16]
    gat_kernel<<<A, 128, 0, stream>>>(nf, adj, node_idx, gat_W, gat_a, gat_Wo,
                                      gat_ao, enc_in);

    // encoder: x = relu(enc_in@W1+b1)@W2+b2
    launch_gemm(enc_in, enc_W1, enc_b1, nullptr, nullptr, t1, A, H, 16, 0, 1, stream);
    launch_gemm(t1, enc_W2, enc_b2, nullptr, nullptr, x,  A, H, H, 0, 0, stream);

    const float scale = 0.03125f;        // 1/sqrt(1024)
    for (int p = 0; p < 4; ++p) {
        // q = h@Wq+bq ; k = (h*ca)@Wk+bk ; v = (h*ca)@Wv+bv
        launch_gemm(h, Wq, bq, nullptr, nullptr, q,    A, 16, H, 0, 0, stream);
        launch_gemm(h, Wk, bk, nullptr, caf,     kbuf, A, 16, H, 0, 0, stream);
        launch_gemm(h, Wv, bv, nullptr, caf,     v,    A, H,  H, 0, 0, stream);
        attn_kernel<<<A, 128, 0, stream>>>(q, kbuf, attn, A, scale);
        // cmsg1 = attn@v ; xc = cmsg1@C_W[p] + C_b[p] + x
        launch_gemm(attn, v, nullptr, nullptr, nullptr, cmsg1, A, H, A, 0, 0, stream);
        launch_gemm(cmsg1, C_W + (size_t)p * H * H, C_b + (size_t)p * H, x, nullptr,
                    xc, A, H, H, 0, 0, stream);
        // gates = xc@Wih^T + bih + h@Whh^T + bhh   (big kernel: N=4096)
        launch_gemm(xc, Wih, bih, nullptr, nullptr, gates, A, 4 * H, H, 1, 0, stream);
        launch_gemm(h,  Whh, bhh, gates,   nullptr, gates, A, 4 * H, H, 1, 0, stream);
        lstm_kernel<<<(A * H + 255) / 256, 256, 0, stream>>>(gates, h, c, A, H);
    }

    // outputs
    copy_kernel<<<(A * H + 255) / 256, 256, 0, stream>>>(h, out_h, A * H);
    launch_gemm(h, vh_W, vh_b, nullptr, nullptr, out_vh, A, 1, H, 0, 0, stream);
    launch_gemm(h, vg_W, vg_b, nullptr, nullptr, out_vg, A, A, H, 0, 0, stream);
    launch_gemm(h, map_W1, map_b1, nullptr, nullptr, map1, A, H, H, 0, 0, stream);
    launch_gemm(map1, map_W2, map_b2, nullptr, nullptr, out_node, A, MAP_OUT, H, 0, 2, stream);
    launch_gemm(h, gnn_W1, gnn_b1, nullptr, nullptr, gnn1, A, H, H, 0, 0, stream);
    launch_gemm(gnn1, gnn_W2, gnn_b2, nullptr, nullptr, out_edge, A, GNN_OUT, H, 0, 2, stream);
}